// MACEDescriptor_71167608095094
// MI455X (gfx1250) — compile-verified
//
#include <hip/hip_runtime.h>
#include <hip/hip_bf16.h>

// MACE descriptor on MI455X (gfx1250), f32 path via V_WMMA_F32_16X16X4_F32.
// Only lm=0..3 of the 16 spherical channels are consumed downstream, so we
// compute only W_mlp4 cols 0..127, ang[:,0:4], and keep the aggregation buffer
// at N*4*64 f32 = 20.5MB (L2-resident; scatter atomics stay on-chip).
//
// Weights are staged in LDS in "pair-major" WMMA-B layout:
//   sw[(p*P + n)*2 + r] = W[2p + r][n]
// so each lane's B operand for v_wmma_f32_16x16x4_f32 is a single aligned
// ds_load_b64 (no VGPR repacking movs). P mod 32 == 16 makes lanes 16..31 hit
// the even banks lanes 0..15 don't -> conflict-free.

#define NNODES 20000
#define NEDGES 200000
#define CDIM   64
#define RDIM   8
#define NELEM  10
#define INV_AVG 0.1f
#define LD     68    // padded A-tile leading dim (bank-friendly, 16B aligned)
#define PW     80    // B pair-stride for N=64  (80 mod 32 == 16)
#define PW4    144   // B pair-stride for N=128 (144 mod 32 == 16)

typedef __attribute__((ext_vector_type(2))) float v2f;
typedef __attribute__((ext_vector_type(8))) float v8f;

__device__ __forceinline__ v8f vzero8() {
  v8f z = {0.f, 0.f, 0.f, 0.f, 0.f, 0.f, 0.f, 0.f};
  return z;
}

// Multi-column-group WMMA: D[cg] += A(16xK) @ B[:, cg*16..+16).
// A tile: row-major in LDS (ld = ldA). Lanes 0-15 take K={k,k+1}, lanes 16-31
// K={k+2,k+3} (ISA 7.12.2 A layout). B: pair-major swizzled (see above); each
// lane loads its {row,row+1} pair as one b64.
template <int K, int NCG>
__device__ __forceinline__ void wmma_mc(const float* As, int ldA,
                                        const float* Bsw, int P,
                                        v8f* acc, int lane) {
  const int row = lane & 15;
  const int col = lane & 15;
  const int kh  = (lane >> 4) << 1;  // 0 or 2
  const int ph  = (lane >> 4);       // pair select within k-group
#pragma unroll
  for (int k = 0; k < K; k += 4) {
    v2f a;
    a.x = As[row * ldA + k + kh];
    a.y = As[row * ldA + k + kh + 1];
    const float* bp = Bsw + (size_t)(((k >> 1) + ph) * P) * 2;
#pragma unroll
    for (int cg = 0; cg < NCG; ++cg) {
      v2f b = *(const v2f*)(bp + (size_t)(col + cg * 16) * 2);
      acc[cg] = __builtin_amdgcn_wmma_f32_16x16x4_f32(false, a, false, b, (short)0,
                                                      acc[cg], false, false);
    }
  }
}

// C/D layout (ISA 7.12.2): VGPR i holds row i (lanes0-15) / row i+8 (lanes16-31).
__device__ __forceinline__ void store_tile(float* dst, int ld, int colBase, v8f v,
                                           int lane) {
  const int col = lane & 15;
  const int rbase = (lane >> 4) << 3;
#pragma unroll
  for (int i = 0; i < 8; ++i) dst[(rbase + i) * ld + colBase + col] = v[i];
}

__device__ __forceinline__ v8f silu8(v8f x) {
#pragma unroll
  for (int i = 0; i < 8; ++i) {
    float v = x[i];
    x[i] = v * (1.0f / (1.0f + __expf(-v)));
  }
  return x;
}

// Cooperative global->LDS copy (row-major, float4 wide).
__device__ __forceinline__ void copy_tile(float* dst, int ldd, const float* src,
                                          int ldsrc, int rows, int cols, int tid,
                                          int nt) {
  const int q = cols >> 2;
  for (int i = tid; i < rows * q; i += nt) {
    const int r = i / q;
    const int c = (i - r * q) << 2;
    *(float4*)(dst + r * ldd + c) = *(const float4*)(src + (size_t)r * ldsrc + c);
  }
}

// Cooperative global->LDS copy into pair-major WMMA-B layout.
// dst[(p*P + n)*2 + r] = src[(2p + r)*ldsrc + n]
__device__ __forceinline__ void copy_tile_swz(float* dst, int P, const float* src,
                                              int ldsrc, int Krows, int cols,
                                              int tid, int nt) {
  const int q = cols >> 2;
  const int pairs = Krows >> 1;
  for (int i = tid; i < pairs * q; i += nt) {
    const int p = i / q;
    const int c = (i - p * q) << 2;
    float4 w0 = *(const float4*)(src + (size_t)(2 * p) * ldsrc + c);
    float4 w1 = *(const float4*)(src + (size_t)(2 * p + 1) * ldsrc + c);
    float* d = dst + (size_t)(p * P + c) * 2;
    d[0] = w0.x; d[1] = w1.x;
    d[2] = w0.y; d[3] = w1.y;
    d[4] = w0.z; d[5] = w1.z;
    d[6] = w0.w; d[7] = w1.w;
  }
}

// ---------------------------------------------------------------- zero scratch
__global__ void k_zero(float* __restrict__ p, int n4) {
  int i = blockIdx.x * blockDim.x + threadIdx.x;
  if (i < n4) ((float4*)p)[i] = make_float4(0.f, 0.f, 0.f, 0.f);
}

// ---------------------------------------------------------------- x = nf @ Wup
__global__ void __launch_bounds__(256) k_xup(const float* __restrict__ nf,
                                             const float* __restrict__ Wup,
                                             float* __restrict__ x) {
  __shared__ float sW[32 * PW * 2];
  __shared__ float sA[8][16 * LD];
  const int tid = threadIdx.x;
  copy_tile_swz(sW, PW, Wup, 64, 64, 64, tid, 256);
  __syncthreads();
  const int wave = tid >> 5, lane = tid & 31;
  const int tile = blockIdx.x * 8 + wave;
  if (tile >= NNODES / 16) return;
  const int nb = tile * 16;
  float* As = sA[wave];
  {
    int r = lane >> 1, c0 = (lane & 1) * 32;
#pragma unroll
    for (int j = 0; j < 32; j += 4)
      *(float4*)(As + r * LD + c0 + j) =
          *(const float4*)(nf + (size_t)(nb + r) * 64 + c0 + j);
  }
  const int col = lane & 15, rbase = (lane >> 4) << 3;
  v8f acc[4];
#pragma unroll
  for (int cg = 0; cg < 4; ++cg) acc[cg] = vzero8();
  wmma_mc<64, 4>(As, LD, sW, PW, acc, lane);
#pragma unroll
  for (int cg = 0; cg < 4; ++cg)
#pragma unroll
    for (int i = 0; i < 8; ++i)
      x[(size_t)(nb + rbase + i) * 64 + cg * 16 + col] = acc[cg][i];
}

// ------------------------------------------------- edge MLP + message scatter
__device__ __forceinline__ void mlp_layer(float* h, const float* Wsw, int lane) {
  v8f acc[4];
#pragma unroll
  for (int cg = 0; cg < 4; ++cg) acc[cg] = vzero8();
  wmma_mc<64, 4>(h, LD, Wsw, PW, acc, lane);
#pragma unroll
  for (int cg = 0; cg < 4; ++cg) store_tile(h, LD, cg * 16, silu8(acc[cg]), lane);
}

__global__ void __launch_bounds__(256) k_edge(
    const float* __restrict__ ang, const float* __restrict__ rad,
    const int* __restrict__ eidx, const float* __restrict__ xbuf,
    const float* __restrict__ W1, const float* __restrict__ W2,
    const float* __restrict__ W3, const float* __restrict__ W4,
    float* __restrict__ agg) {
  __shared__ float sW1[4 * PW * 2];    // K=8  -> 4 pairs
  __shared__ float sW2[32 * PW * 2];
  __shared__ float sW3[32 * PW * 2];
  __shared__ float sW4[32 * PW4 * 2];  // N=128 (only first 128 of 256 cols used)
  __shared__ float sXS[8][16 * LD];
  __shared__ float sH[8][16 * LD];
  __shared__ float sR[8][16 * 8];
  __shared__ float sAng[8][16 * 4];
  __shared__ int   sSnd[8][16];
  __shared__ int   sRcv[8][16];
  const int tid = threadIdx.x;
  copy_tile_swz(sW1, PW, W1, 64, RDIM, 64, tid, 256);
  copy_tile_swz(sW2, PW, W2, 64, 64, 64, tid, 256);
  copy_tile_swz(sW3, PW, W3, 64, 64, 64, tid, 256);
  copy_tile_swz(sW4, PW4, W4, 256, 64, 128, tid, 256);
  __syncthreads();
  const int wave = tid >> 5, lane = tid & 31;
  const int tile = blockIdx.x * 8 + wave;
  if (tile >= NEDGES / 16) return;
  const int eb = tile * 16;
  float* xs = sXS[wave];
  float* h  = sH[wave];
  float* rT = sR[wave];
  float* aT = sAng[wave];
  int* sn = sSnd[wave];
  int* rc = sRcv[wave];
  // stage per-edge data (per-wave LDS slice; DS ops are in-order within a wave)
  if (lane < 16) {
    sn[lane] = eidx[eb + lane];
    rc[lane] = eidx[NEDGES + eb + lane];
    *(float4*)(aT + lane * 4) = *(const float4*)(ang + (size_t)(eb + lane) * 16);
  }
  *(float4*)(rT + lane * 4) = *(const float4*)(rad + (size_t)eb * 8 + lane * 4);
  {
    int r = lane >> 1, c0 = (lane & 1) * 32;
    int s = sn[r];
#pragma unroll
    for (int j = 0; j < 32; j += 4)
      *(float4*)(xs + r * LD + c0 + j) =
          *(const float4*)(xbuf + (size_t)s * 64 + c0 + j);
  }
  const int col = lane & 15, rbase = (lane >> 4) << 3;
  // h1 = silu(radial @ W_mlp1), K = 8
  {
    v8f acc[4];
#pragma unroll
    for (int cg = 0; cg < 4; ++cg) acc[cg] = vzero8();
    wmma_mc<8, 4>(rT, 8, sW1, PW, acc, lane);
#pragma unroll
    for (int cg = 0; cg < 4; ++cg)
      store_tile(h, LD, cg * 16, silu8(acc[cg]), lane);
  }
  mlp_layer(h, sW2, lane);  // h2
  mlp_layer(h, sW3, lane);  // h3
  // tp (cols 0..127) + fused message scatter: msg = x[snd]*ang*tp/AVG_NEI
  v8f t[8];
#pragma unroll
  for (int cg = 0; cg < 8; ++cg) t[cg] = vzero8();
  wmma_mc<64, 8>(h, LD, sW4, PW4, t, lane);
#pragma unroll
  for (int cg = 0; cg < 8; ++cg) {
    const int c = (cg & 3) * 16 + col;
    if (cg < 4) {  // l=0 -> lm=0
#pragma unroll
      for (int i = 0; i < 8; ++i) {
        const int rr = rbase + i;
        const float v = t[cg][i] * xs[rr * LD + c] * INV_AVG;
        atomicAdd(agg + (size_t)rc[rr] * 256 + c, v * aT[rr * 4 + 0]);
      }
    } else {  // l=1 -> lm=1,2,3
#pragma unroll
      for (int i = 0; i < 8; ++i) {
        const int rr = rbase + i;
        const float v = t[cg][i] * xs[rr * LD + c] * INV_AVG;
#pragma unroll
        for (int lm = 1; lm < 4; ++lm)
          atomicAdd(agg + (size_t)rc[rr] * 256 + lm * 64 + c, v * aT[rr * 4 + lm]);
      }
    }
  }
}

// ---------------------------------------------- node output (sc0/feats/proj)
__global__ void __launch_bounds__(256) k_nodeout(
    const float* __restrict__ nf, const float* __restrict__ oh,
    const float* __restrict__ agg, const float* __restrict__ Wlin,
    const float* __restrict__ Wskip, const float* __restrict__ Wc0,
    const float* __restrict__ Wc1, const float* __restrict__ Wp0,
    const float* __restrict__ Wp1, float* __restrict__ out) {
  __shared__ float sL0[32 * PW * 2];
  __shared__ float sL1[32 * PW * 2];
  __shared__ float sP0[32 * PW * 2];
  __shared__ float sP1[32 * PW * 2];
  __shared__ float sC0[NELEM * 64];
  __shared__ float sC1[NELEM * 64];
  __shared__ float sSk[32 * PW * 2];
  __shared__ float sNF[8][16 * LD];
  __shared__ float sOH[8][16 * NELEM];
  __shared__ float sWk[8][16 * LD];
  const int tid = threadIdx.x;
  copy_tile_swz(sL0, PW, Wlin, 64, 64, 64, tid, 256);
  copy_tile_swz(sL1, PW, Wlin + 64 * 64, 64, 64, 64, tid, 256);
  copy_tile_swz(sP0, PW, Wp0, 64, 64, 64, tid, 256);
  copy_tile_swz(sP1, PW, Wp1, 64, 64, 64, tid, 256);
  copy_tile(sC0, 64, Wc0, 64, NELEM, 64, tid, 256);
  copy_tile(sC1, 64, Wc1, 64, NELEM, 64, tid, 256);
  __syncthreads();
  const int wave = tid >> 5, lane = tid & 31;
  const int tile = blockIdx.x * 8 + wave;
  const bool valid = tile < NNODES / 16;
  const int nb = tile * 16;
  float* nfT = sNF[wave];
  float* ohT = sOH[wave];
  float* wk  = sWk[wave];
  const int col = lane & 15, rbase = (lane >> 4) << 3;
  if (valid) {
    int r = lane >> 1, c0 = (lane & 1) * 32;
#pragma unroll
    for (int j = 0; j < 32; j += 4)
      *(float4*)(nfT + r * LD + c0 + j) =
          *(const float4*)(nf + (size_t)(nb + r) * 64 + c0 + j);
    if (lane < 16) {
#pragma unroll
      for (int a = 0; a < NELEM; ++a)
        ohT[lane * NELEM + a] = oh[(size_t)(nb + lane) * NELEM + a];
    }
  }
  // sc0[n,k] = sum_a one_hot[n,a] * (nf @ W_skip[:,a,:])[n,k], streamed per a
  v8f sc[4];
#pragma unroll
  for (int cg = 0; cg < 4; ++cg) sc[cg] = vzero8();
  for (int a = 0; a < NELEM; ++a) {
    __syncthreads();  // previous slice fully consumed
    copy_tile_swz(sSk, PW, Wskip + a * 64, NELEM * 64, 64, 64, tid, 256);
    if (a + 1 < NELEM)  // gfx1250 global_prefetch of next slice into caches
      __builtin_prefetch(Wskip + (a + 1) * 64 + (size_t)(tid & 63) * (NELEM * 64),
                         0, 1);
    __syncthreads();
    if (valid) {
      v8f t[4];
#pragma unroll
      for (int cg = 0; cg < 4; ++cg) t[cg] = vzero8();
      wmma_mc<64, 4>(nfT, LD, sSk, PW, t, lane);
#pragma unroll
      for (int cg = 0; cg < 4; ++cg)
#pragma unroll
        for (int i = 0; i < 8; ++i)
          sc[cg][i] += ohT[(rbase + i) * NELEM + a] * t[cg][i];
    }
  }
  if (!valid) return;
  // per-channel: feats_m = agg_m @ W_lin[l(m)], scale by wn, project
#pragma unroll
  for (int m = 0; m < 4; ++m) {
    {
      int r = lane >> 1, c0 = (lane & 1) * 32;
#pragma unroll
      for (int j = 0; j < 32; j += 4)
        *(float4*)(wk + r * LD + c0 + j) =
            *(const float4*)(agg + (size_t)(nb + r) * 256 + m * 64 + c0 + j);
    }
    const float* Wl = (m == 0) ? sL0 : sL1;
    const float* Wc = (m == 0) ? sC0 : sC1;
    v8f f[4];
#pragma unroll
    for (int cg = 0; cg < 4; ++cg) f[cg] = vzero8();
    wmma_mc<64, 4>(wk, LD, Wl, PW, f, lane);
    // scale rows by wn (one_hot @ Wc), write back to wk for projection
#pragma unroll
    for (int cg = 0; cg < 4; ++cg) {
      const int cc = cg * 16 + col;
#pragma unroll
      for (int i = 0; i < 8; ++i) {
        const int rr = rbase + i;
        float wn = 0.f;
#pragma unroll
        for (int a = 0; a < NELEM; ++a)
          wn += ohT[rr * NELEM + a] * Wc[a * 64 + cc];
        wk[rr * LD + cc] = f[cg][i] * wn;
      }
    }
    v8f o[4];
    if (m == 0) {
#pragma unroll
      for (int cg = 0; cg < 4; ++cg) o[cg] = sc[cg];
      wmma_mc<64, 4>(wk, LD, sP0, PW, o, lane);
#pragma unroll
      for (int cg = 0; cg < 4; ++cg)
#pragma unroll
        for (int i = 0; i < 8; ++i)
          out[(size_t)(nb + rbase + i) * 256 + cg * 16 + col] = o[cg][i];
    } else {
#pragma unroll
      for (int cg = 0; cg < 4; ++cg) o[cg] = vzero8();
      wmma_mc<64, 4>(wk, LD, sP1, PW, o, lane);
#pragma unroll
      for (int cg = 0; cg < 4; ++cg)
#pragma unroll
        for (int i = 0; i < 8; ++i)
          out[(size_t)(nb + rbase + i) * 256 + 64 + (cg * 16 + col) * 3 + (m - 1)] =
              o[cg][i];
    }
  }
}

extern "C" void kernel_launch(void* const* d_in, const int* in_sizes, int n_in,
                              void* d_out, int out_size, void* d_ws, size_t ws_size,
                              hipStream_t stream) {
  const float* nf   = (const float*)d_in[0];
  const float* oh   = (const float*)d_in[1];
  const float* ang  = (const float*)d_in[2];
  const float* rad  = (const float*)d_in[3];
  const float* Wup  = (const float*)d_in[4];
  const float* W1   = (const float*)d_in[5];
  const float* W2   = (const float*)d_in[6];
  const float* W3   = (const float*)d_in[7];
  const float* W4   = (const float*)d_in[8];
  const float* Wlin = (const float*)d_in[9];
  const float* Wsk  = (const float*)d_in[10];
  const float* Wc0  = (const float*)d_in[11];
  const float* Wc1  = (const float*)d_in[12];
  const float* Wp0  = (const float*)d_in[13];
  const float* Wp1  = (const float*)d_in[14];
  const int*   eidx = (const int*)d_in[15];
  float* out = (float*)d_out;

  float* x   = (float*)d_ws;             // N*64 f32
  float* agg = x + (size_t)NNODES * 64;  // N*4*64 f32 (L2-resident)

  const int n4 = NNODES * 256 / 4;
  k_zero<<<(n4 + 255) / 256, 256, 0, stream>>>(agg, n4);
  k_xup<<<(NNODES / 16 + 7) / 8, 256, 0, stream>>>(nf, Wup, x);
  k_edge<<<(NEDGES / 16 + 7) / 8, 256, 0, stream>>>(ang, rad, eidx, x, W1, W2, W3,
                                                    W4, agg);
  k_nodeout<<<(NNODES / 16 + 7) / 8, 256, 0, stream>>>(nf, oh, agg, Wlin, Wsk, Wc0,
                                                       Wc1, Wp0, Wp1, out);
}